// MaxMarginLoss_69415261438244
// MI455X (gfx1250) — compile-verified
//
#include <hip/hip_runtime.h>
#include <math.h>

typedef float v2f __attribute__((ext_vector_type(2)));
typedef float v4f __attribute__((ext_vector_type(4)));
typedef float v8f __attribute__((ext_vector_type(8)));

#define Bsz 4
#define Ssz 512
#define Dsz 512
#define Vsz 32000
#define ROWS (Bsz * Ssz)          // 2048
#define GAMMA 0.5f
#define EPSN 1e-12f
#define NSLICE 10
#define SLICE_V (Vsz / NSLICE)    // 3200 vocab entries per slice
#define NCAND (NSLICE * 4)        // 40 candidates per row
#define NEG_BIG (-3.402823466e38f)

// ---------------- init scalars ----------------
__global__ __launch_bounds__(64) void init_kernel(float* sumLoss, int* cnt) {
  if (threadIdx.x == 0) { *sumLoss = 0.0f; *cnt = 0; }
}

// ---------------- 1/max(||emb_v||, eps) ----------------
__global__ __launch_bounds__(256) void vocnorm_kernel(const float* __restrict__ emb,
                                                      float* __restrict__ invn) {
  int wave = threadIdx.x >> 5, lane = threadIdx.x & 31;
  int v = blockIdx.x * 8 + wave;
  const float* e = emb + (size_t)v * Dsz;
  float ss = 0.0f;
#pragma unroll
  for (int i = 0; i < Dsz / 32; ++i) { float x = e[lane + 32 * i]; ss += x * x; }
#pragma unroll
  for (int off = 16; off >= 1; off >>= 1) ss += __shfl_xor(ss, off, 32);
  if (lane == 0) invn[v] = 1.0f / fmaxf(sqrtf(ss), EPSN);
}

// ---------------- out_norm rows and q = out_norm - emb[target] ----------------
__global__ __launch_bounds__(256) void rows_kernel(const float* __restrict__ preds,
                                                   const float* __restrict__ emb,
                                                   const long long* __restrict__ target,
                                                   const int* __restrict__ padp,
                                                   float* __restrict__ outn,
                                                   float* __restrict__ qbuf,
                                                   int* __restrict__ cnt) {
  int wave = threadIdx.x >> 5, lane = threadIdx.x & 31;
  int row = blockIdx.x * 8 + wave;         // 0..2047
  int b = row >> 9, s = row & (Ssz - 1);
  float xr[16];
  float ss = 0.0f;
#pragma unroll
  for (int i = 0; i < 16; ++i) {
    int d = lane + 32 * i;
    float x = preds[((size_t)(b * Dsz + d)) * Ssz + s];  // preds[b][d][s]
    xr[i] = x; ss += x * x;
  }
#pragma unroll
  for (int off = 16; off >= 1; off >>= 1) ss += __shfl_xor(ss, off, 32);
  float inv = 1.0f / fmaxf(sqrtf(ss), EPSN);
  int tgt = (int)target[row];
  const float* te = emb + (size_t)tgt * Dsz;
  float* on = outn + (size_t)row * Dsz;
  float* qr = qbuf + (size_t)row * Dsz;
#pragma unroll
  for (int i = 0; i < 16; ++i) {
    int d = lane + 32 * i;
    float o = xr[i] * inv;
    on[d] = o;
    qr[d] = o - te[d];
  }
  if (lane == 0 && tgt != *padp) atomicAdd(cnt, 1);
}

// ---------------- main WMMA GEMM + running argmax ----------------
// grid (ROWS/32, NSLICE), block 256 (8 waves = 2 row-tiles x 4 col-pairs)
// Each wave owns two 16x16 output tiles (independent WMMA chains).
// K processed in groups of 8 with column sets {8g..8g+1, 8g+4..8g+5} /
// {8g+2..8g+3, 8g+6..8g+7} so each lane's two-WMMA A (and B) data is one b128.
// Register double-buffering: group g+1 fragments are loaded before group g's WMMAs.
__global__ __launch_bounds__(256) void argmax_kernel(const float* __restrict__ emb,
                                                     const float* __restrict__ qbuf,
                                                     const float* __restrict__ invn,
                                                     float* __restrict__ candV,
                                                     int* __restrict__ candI) {
  __shared__ float As[32 * 512];  // 64 KB, XOR-swizzled in float4 units
  int rowBlock = blockIdx.x;      // 0..63 (32 rows each)
  int slice = blockIdx.y;         // 0..9
  int t = threadIdx.x;

  // stage 32 q-rows into LDS; float4 unit c4 stored at (c4 ^ (row&15))
  const float* qblk = qbuf + (size_t)rowBlock * 32 * Dsz;
  const v4f* qv4 = (const v4f*)qblk;
  v4f* av4 = (v4f*)As;
#pragma unroll
  for (int i = 0; i < 16; ++i) {
    int idx = t + 256 * i;            // 0..4095 float4 units
    int row = idx >> 7;               // /128
    int c4 = idx & 127;
    av4[row * 128 + (c4 ^ (row & 15))] = qv4[row * 128 + c4];
  }
  __syncthreads();

  int lane = t & 31;
  int wave = t >> 5;
  int r = wave & 1;        // row tile within 32-row block
  int c = wave >> 1;       // col pair 0..3 (32 vocab cols each)
  int half = lane >> 4;
  int n16 = lane & 15;
  int arow = r * 16 + n16;
  const float* aRow = As + arow * 512;

  // per-lane A-fragment LDS offsets (floats), reused every block of 8 groups
  int aOff[8];
#pragma unroll
  for (int gi = 0; gi < 8; ++gi) aOff[gi] = 4 * ((2 * gi + half) ^ n16);

  float bestV0[8], bestV1[8]; int bestI0[8], bestI1[8];
#pragma unroll
  for (int m = 0; m < 8; ++m) {
    bestV0[m] = NEG_BIG; bestI0[m] = 0;
    bestV1[m] = NEG_BIG; bestI1[m] = 0;
  }

  for (int it = 0; it < SLICE_V / 128; ++it) {         // 25 iterations
    int vcol0 = slice * SLICE_V + it * 128 + c * 32 + n16;
    int vcol1 = vcol0 + 16;
    const float* bptr0 = emb + (size_t)vcol0 * Dsz + 4 * half;
    const float* bptr1 = emb + (size_t)vcol1 * Dsz + 4 * half;
    float inv0 = invn[vcol0];
    float inv1 = invn[vcol1];
    if (it + 1 < SLICE_V / 128) {                      // speculative next-tile prefetch
      __builtin_prefetch(bptr0 + 128 * Dsz, 0, 0);
      __builtin_prefetch(bptr1 + 128 * Dsz, 0, 0);
    }
    v8f acc0 = {0.f, 0.f, 0.f, 0.f, 0.f, 0.f, 0.f, 0.f};
    v8f acc1 = {0.f, 0.f, 0.f, 0.f, 0.f, 0.f, 0.f, 0.f};

    // prologue: group 0 fragments
    v4f a4 = *(const v4f*)(aRow + aOff[0]);
    v4f b40 = *(const v4f*)(bptr0);
    v4f b41 = *(const v4f*)(bptr1);
#pragma unroll
    for (int g = 0; g < Dsz / 8 - 1; ++g) {            // 63 steady-state groups
      int gn = g + 1;
      v4f an = *(const v4f*)(aRow + 64 * (gn >> 3) + aOff[gn & 7]);
      v4f nb0 = *(const v4f*)(bptr0 + 8 * gn);
      v4f nb1 = *(const v4f*)(bptr1 + 8 * gn);
      v2f a0 = {a4.x, a4.y}, a1 = {a4.z, a4.w};
      v2f b00 = {b40.x, b40.y}, b01 = {b40.z, b40.w};
      v2f b10 = {b41.x, b41.y}, b11 = {b41.z, b41.w};
      acc0 = __builtin_amdgcn_wmma_f32_16x16x4_f32(false, a0, false, b00, (short)0, acc0, false, false);
      acc1 = __builtin_amdgcn_wmma_f32_16x16x4_f32(false, a0, false, b10, (short)0, acc1, false, false);
      acc0 = __builtin_amdgcn_wmma_f32_16x16x4_f32(false, a1, false, b01, (short)0, acc0, false, false);
      acc1 = __builtin_amdgcn_wmma_f32_16x16x4_f32(false, a1, false, b11, (short)0, acc1, false, false);
      a4 = an; b40 = nb0; b41 = nb1;
    }
    {  // epilogue: last group
      v2f a0 = {a4.x, a4.y}, a1 = {a4.z, a4.w};
      v2f b00 = {b40.x, b40.y}, b01 = {b40.z, b40.w};
      v2f b10 = {b41.x, b41.y}, b11 = {b41.z, b41.w};
      acc0 = __builtin_amdgcn_wmma_f32_16x16x4_f32(false, a0, false, b00, (short)0, acc0, false, false);
      acc1 = __builtin_amdgcn_wmma_f32_16x16x4_f32(false, a0, false, b10, (short)0, acc1, false, false);
      acc0 = __builtin_amdgcn_wmma_f32_16x16x4_f32(false, a1, false, b01, (short)0, acc0, false, false);
      acc1 = __builtin_amdgcn_wmma_f32_16x16x4_f32(false, a1, false, b11, (short)0, acc1, false, false);
    }

#pragma unroll
    for (int m = 0; m < 8; ++m) {
      float sc0 = acc0[m] * inv0;
      float sc1 = acc1[m] * inv1;
      if (sc0 > bestV0[m]) { bestV0[m] = sc0; bestI0[m] = vcol0; }  // strict > keeps first index
      if (sc1 > bestV1[m]) { bestV1[m] = sc1; bestI1[m] = vcol1; }
    }
  }

  // merge the wave's two tiles, reduce across the 16 lanes of each half, emit candidates
#pragma unroll
  for (int m = 0; m < 8; ++m) {
    float v = bestV0[m]; int i = bestI0[m];
    if (bestV1[m] > v || (bestV1[m] == v && bestI1[m] < i)) { v = bestV1[m]; i = bestI1[m]; }
#pragma unroll
    for (int off = 8; off >= 1; off >>= 1) {
      float ov = __shfl_xor(v, off, 32);
      int oi = __shfl_xor(i, off, 32);
      if (ov > v || (ov == v && oi < i)) { v = ov; i = oi; }
    }
    if ((lane & 15) == 0) {
      int row = rowBlock * 32 + r * 16 + m + 8 * half;
      int slot = slice * 4 + c;
      candV[row * NCAND + slot] = v;
      candI[row * NCAND + slot] = i;
    }
  }
}

// ---------------- combine candidates, cosine epilogue, hinge ----------------
__global__ __launch_bounds__(256) void finalize_kernel(const float* __restrict__ emb,
                                                       const float* __restrict__ outn,
                                                       const float* __restrict__ invn,
                                                       const float* __restrict__ candV,
                                                       const int* __restrict__ candI,
                                                       const long long* __restrict__ target,
                                                       const int* __restrict__ padp,
                                                       float* __restrict__ sumLoss) {
  int wave = threadIdx.x >> 5, lane = threadIdx.x & 31;
  int row = blockIdx.x * 8 + wave;
  float bv = NEG_BIG; int bi = 0x7fffffff;
  for (int j = lane; j < NCAND; j += 32) {
    float v = candV[row * NCAND + j]; int i = candI[row * NCAND + j];
    if (v > bv || (v == bv && i < bi)) { bv = v; bi = i; }
  }
#pragma unroll
  for (int off = 16; off >= 1; off >>= 1) {
    float ov = __shfl_xor(bv, off, 32);
    int oi = __shfl_xor(bi, off, 32);
    if (ov > bv || (ov == bv && oi < bi)) { bv = ov; bi = oi; }
  }
  int tgt = (int)target[row];
  const float* on = outn + (size_t)row * Dsz;
  const float* ej = emb + (size_t)bi * Dsz;
  const float* et = emb + (size_t)tgt * Dsz;
  float dj = 0.0f, dt = 0.0f;
#pragma unroll
  for (int i = 0; i < Dsz / 32; ++i) {
    int d = lane + 32 * i;
    float o = on[d];
    dj += o * ej[d];
    dt += o * et[d];
  }
#pragma unroll
  for (int off = 16; off >= 1; off >>= 1) {
    dj += __shfl_xor(dj, off, 32);
    dt += __shfl_xor(dt, off, 32);
  }
  if (lane == 0) {
    float cosJ = dj * invn[bi];
    float cosT = dt * invn[tgt];
    float diff = fmaxf(GAMMA + cosJ - cosT, 0.0f);
    if (tgt != *padp) atomicAdd(sumLoss, diff);
  }
}

__global__ __launch_bounds__(64) void div_kernel(const float* sumLoss, const int* cnt,
                                                 float* out) {
  if (threadIdx.x == 0) out[0] = sumLoss[0] / (float)cnt[0];
}

extern "C" void kernel_launch(void* const* d_in, const int* in_sizes, int n_in,
                              void* d_out, int out_size, void* d_ws, size_t ws_size,
                              hipStream_t stream) {
  const float* preds = (const float*)d_in[0];
  const float* emb = (const float*)d_in[1];
  const long long* target = (const long long*)d_in[2];  // int64 targets
  const int* padp = (const int*)d_in[3];                // pad_id scalar (low word)
  float* out = (float*)d_out;

  float* outn = (float*)d_ws;                           // [2048][512]
  float* qbuf = outn + (size_t)ROWS * Dsz;              // [2048][512]
  float* invn = qbuf + (size_t)ROWS * Dsz;              // [32000]
  float* candV = invn + Vsz;                            // [2048][40]
  int* candI = (int*)(candV + (size_t)ROWS * NCAND);    // [2048][40]
  float* sumLoss = (float*)(candI + (size_t)ROWS * NCAND);
  int* cnt = (int*)(sumLoss + 1);

  init_kernel<<<1, 64, 0, stream>>>(sumLoss, cnt);
  vocnorm_kernel<<<Vsz / 8, 256, 0, stream>>>(emb, invn);
  rows_kernel<<<ROWS / 8, 256, 0, stream>>>(preds, emb, target, padp, outn, qbuf, cnt);
  argmax_kernel<<<dim3(ROWS / 32, NSLICE), 256, 0, stream>>>(emb, qbuf, invn, candV, candI);
  finalize_kernel<<<ROWS / 8, 256, 0, stream>>>(emb, outn, invn, candV, candI, target,
                                                padp, sumLoss);
  div_kernel<<<1, 64, 0, stream>>>(sumLoss, cnt, out);
}